// GNNModule_85074712199473
// MI455X (gfx1250) — compile-verified
//
#include <hip/hip_runtime.h>

// CDNA5 / gfx1250: wave32, WMMA f32 16x16x4 for all GEMMs (full fp32 precision).
// - B matrices pre-packed into WMMA fragment order (lane-contiguous b128 loads).
// - P/Q GEMMs fused (shared A reads), aggregation GEMM fused with bias+LayerNorm
//   +residual+LeakyReLU epilogue done in-register with wave32 shuffle butterflies.

typedef __attribute__((ext_vector_type(2))) float v2f;
typedef __attribute__((ext_vector_type(8))) float v8f;

#define GNN_SLOPE 0.01f
#define GNN_EPS   1e-5f
#define GNN_BSN   16
#define GNN_N     2000
#define GNN_E     32000
#define GNN_NT    (GNN_BSN * GNN_N)   // 32000 total nodes
#define GNN_DOUT  128

__device__ __forceinline__ float lrelu(float v) { return v >= 0.0f ? v : GNN_SLOPE * v; }

#define WMMA_F32(A, B, C) \
    __builtin_amdgcn_wmma_f32_16x16x4_f32(false, (A), false, (B), (short)0, (C), false, false)

// ---------------------------------------------------------------- utilities
__global__ __launch_bounds__(256) void k_fill_zero(float* __restrict__ p, int n) {
    int i = blockIdx.x * 256 + threadIdx.x;
    if (i < n) p[i] = 0.0f;
}

// Pack B[K x 128] (row-major) into WMMA fragment order:
//   Bp[kk*512 + lane*16 + t*2 + comp] = B[(kk*4 + 2*(lane>>4) + comp)*128 + t*16 + (lane&15)]
// so one lane's B data for one k-step (8 n-tiles) is 64 contiguous bytes.
// DIFF variant packs (W1_top - W1_bot) for the concat-GEMM refactor.
template <bool DIFF>
__global__ __launch_bounds__(256) void k_pack_b(const float* __restrict__ B,
                                                float* __restrict__ Bp, int K) {
    int i = blockIdx.x * 256 + threadIdx.x;          // 0 .. K*128-1
    const int kk   = i >> 9;
    const int rem  = i & 511;
    const int lane = rem >> 4;
    const int j    = rem & 15;
    const int t    = j >> 1;
    const int comp = j & 1;
    const int r    = kk * 4 + 2 * (lane >> 4) + comp;
    const int c    = t * 16 + (lane & 15);
    const int src  = r * GNN_DOUT + c;
    float v = B[src];
    if (DIFF) v -= B[K * GNN_DOUT + src];
    Bp[i] = v;
}

// ---------------------------------------------------------------- fused P/Q GEMM
// P = A @ Wa, Q = A @ Wb  (A read once).  One wave -> 16 rows x 128 cols per
// output, dual accumulator sets.  8 waves/block -> 128 rows; grid = 250 exact.
template <int K>
__global__ __launch_bounds__(256) void k_gemm_pq(const float* __restrict__ A,
                                                 const float* __restrict__ BpA,
                                                 const float* __restrict__ BpB,
                                                 float* __restrict__ P,
                                                 float* __restrict__ Q) {
    const int wave = threadIdx.x >> 5;
    const int lane = threadIdx.x & 31;
    const int half = lane >> 4;
    const int lm   = lane & 15;
    const int m0   = blockIdx.x * 128 + wave * 16;

    v8f accP[8] = {};
    v8f accQ[8] = {};

    const float*  arow = A + (size_t)(m0 + lm) * K + 2 * half;
    const float4* bpA  = (const float4*)(BpA + (size_t)lane * 16);
    const float4* bpB  = (const float4*)(BpB + (size_t)lane * 16);

#pragma unroll
    for (int k = 0; k < K; k += 4) {
        const v2f a = *(const v2f*)(arow + k);
        const float4* ba = bpA + (size_t)(k >> 2) * 128;
        const float4* bb = bpB + (size_t)(k >> 2) * 128;
        float4 pa[4], pb[4];
#pragma unroll
        for (int i = 0; i < 4; ++i) { pa[i] = ba[i]; pb[i] = bb[i]; }
        v2f b;
#pragma unroll
        for (int i = 0; i < 4; ++i) {
            b.x = pa[i].x; b.y = pa[i].y; accP[2 * i + 0] = WMMA_F32(a, b, accP[2 * i + 0]);
            b.x = pa[i].z; b.y = pa[i].w; accP[2 * i + 1] = WMMA_F32(a, b, accP[2 * i + 1]);
            b.x = pb[i].x; b.y = pb[i].y; accQ[2 * i + 0] = WMMA_F32(a, b, accQ[2 * i + 0]);
            b.x = pb[i].z; b.y = pb[i].w; accQ[2 * i + 1] = WMMA_F32(a, b, accQ[2 * i + 1]);
        }
    }

    // C/D layout: VGPR r -> row m0 + r + 8*half, col t*16 + lm
    const size_t rb = (size_t)(m0 + 8 * half) * GNN_DOUT + lm;
    float* prow = P + rb;
    float* qrow = Q + rb;
#pragma unroll
    for (int r = 0; r < 8; ++r)
#pragma unroll
        for (int t = 0; t < 8; ++t) {
            prow[(size_t)r * GNN_DOUT + t * 16] = accP[t][r];
            qrow[(size_t)r * GNN_DOUT + t * 16] = accQ[t][r];
        }
}

// ---------------------------------------------------------------- edge stage
// t = lrelu(P[dst] + Q[src] + b1) accumulated into S[dst] (L2-resident atomics),
// cnt[dst] += 1.  One lane per (edge, 4 channels).  grid = (E*32/256, BSN).
__global__ __launch_bounds__(256) void k_edge(const float* __restrict__ P,
                                              const float* __restrict__ Q,
                                              const float* __restrict__ b1,
                                              const int* __restrict__ ei,
                                              float* __restrict__ S,
                                              float* __restrict__ cnt) {
    const int t   = blockIdx.x * 256 + threadIdx.x;   // 0 .. E*32-1
    const int e   = t >> 5;
    const int c   = (t & 31) * 4;
    const int off = blockIdx.y * GNN_N;               // per-graph node offset
    const int src = ei[e] + off;                      // edge_index[0] = j
    const int dst = ei[GNN_E + e] + off;              // edge_index[1] = i

    const float4 p  = *(const float4*)(P + (size_t)dst * GNN_DOUT + c);
    const float4 q  = *(const float4*)(Q + (size_t)src * GNN_DOUT + c);
    const float4 bb = *(const float4*)(b1 + c);

    float* sp = S + (size_t)dst * GNN_DOUT + c;
    atomicAdd(sp + 0, lrelu(p.x + q.x + bb.x));
    atomicAdd(sp + 1, lrelu(p.y + q.y + bb.y));
    atomicAdd(sp + 2, lrelu(p.z + q.z + bb.z));
    atomicAdd(sp + 3, lrelu(p.w + q.w + bb.w));
    if ((t & 31) == 0) atomicAdd(cnt + dst, 1.0f);
}

// ------------------------------------------- aggregation GEMM + LN (fused)
// out = lrelu( LN( (S/max(cnt,1)) @ W2 + [cnt>0]*b2 ) * g + beta (+ xres) )
// The 16x128 wave tile holds complete rows in registers: acc[t][r] over t spans
// all 128 cols of row m0+r+8*half, lanes {0..15} / {16..31} are independent
// 16-lane groups -> LayerNorm via 4-step __shfl_xor butterflies, no extra pass.
template <bool RES>
__global__ __launch_bounds__(256) void k_gemm_agg_ln(const float* __restrict__ S,
                                                     const float* __restrict__ Bp2,
                                                     const float* __restrict__ cnt,
                                                     const float* __restrict__ b2,
                                                     const float* __restrict__ g,
                                                     const float* __restrict__ beta,
                                                     const float* __restrict__ xres,
                                                     float* __restrict__ out) {
    const int wave = threadIdx.x >> 5;
    const int lane = threadIdx.x & 31;
    const int half = lane >> 4;
    const int lm   = lane & 15;
    const int m0   = blockIdx.x * 128 + wave * 16;

    const float cntv  = cnt[m0 + lm];                 // lane L holds cnt[m0 + (L&15)]
    const float scale = 1.0f / fmaxf(cntv, 1.0f);     // mean aggregation

    v8f acc[8] = {};
    const float*  arow = S + (size_t)(m0 + lm) * 128 + 2 * half;
    const float4* bp   = (const float4*)(Bp2 + (size_t)lane * 16);

#pragma unroll
    for (int k = 0; k < 128; k += 4) {
        v2f a = *(const v2f*)(arow + k);
        a.x *= scale; a.y *= scale;
        const float4* bk = bp + (size_t)(k >> 2) * 128;
        float4 pb[4];
#pragma unroll
        for (int i = 0; i < 4; ++i) pb[i] = bk[i];
        v2f b;
#pragma unroll
        for (int i = 0; i < 4; ++i) {
            b.x = pb[i].x; b.y = pb[i].y; acc[2 * i + 0] = WMMA_F32(a, b, acc[2 * i + 0]);
            b.x = pb[i].z; b.y = pb[i].w; acc[2 * i + 1] = WMMA_F32(a, b, acc[2 * i + 1]);
        }
    }

    // Per-lane column parameters (col = t*16 + lm, shared across all 8 rows).
    float b2v[8], gv[8], bev[8];
#pragma unroll
    for (int t = 0; t < 8; ++t) {
        b2v[t] = b2[t * 16 + lm];
        gv[t]  = g[t * 16 + lm];
        bev[t] = beta[t * 16 + lm];
    }

    const size_t rb = (size_t)(m0 + 8 * half) * GNN_DOUT + lm;
    float* crow = out + rb;
    const float* rrow = RES ? (xres + rb) : nullptr;

#pragma unroll
    for (int r = 0; r < 8; ++r) {
        // bias only for nodes with >=1 edge (reference adds b2 per edge message)
        const float cr   = __shfl(cntv, r + 8 * half, 32);  // cnt of this row
        const float hasE = (cr > 0.0f) ? 1.0f : 0.0f;

        float h[8];
        float s = 0.0f;
#pragma unroll
        for (int t = 0; t < 8; ++t) { h[t] = acc[t][r] + hasE * b2v[t]; s += h[t]; }
#pragma unroll
        for (int m = 8; m >= 1; m >>= 1) s += __shfl_xor(s, m, 32);   // within 16-lane group
        const float mu = s * (1.0f / GNN_DOUT);

        float s2 = 0.0f;
#pragma unroll
        for (int t = 0; t < 8; ++t) { h[t] -= mu; s2 += h[t] * h[t]; }
#pragma unroll
        for (int m = 8; m >= 1; m >>= 1) s2 += __shfl_xor(s2, m, 32);
        const float rstd = rsqrtf(s2 * (1.0f / GNN_DOUT) + GNN_EPS);

#pragma unroll
        for (int t = 0; t < 8; ++t) {
            float o = h[t] * rstd * gv[t] + bev[t];
            if (RES) o += rrow[(size_t)r * GNN_DOUT + t * 16];
            crow[(size_t)r * GNN_DOUT + t * 16] = lrelu(o);
        }
    }
}

// ---------------------------------------------------------------- driver
extern "C" void kernel_launch(void* const* d_in, const int* in_sizes, int n_in,
                              void* d_out, int out_size, void* d_ws, size_t ws_size,
                              hipStream_t stream) {
    (void)in_sizes; (void)n_in; (void)out_size; (void)ws_size;

    const float* x  = (const float*)d_in[0];
    const int*   ei = (const int*)d_in[1];
    const float* W1[2] = {(const float*)d_in[2],  (const float*)d_in[8]};
    const float* b1[2] = {(const float*)d_in[3],  (const float*)d_in[9]};
    const float* W2[2] = {(const float*)d_in[4],  (const float*)d_in[10]};
    const float* b2[2] = {(const float*)d_in[5],  (const float*)d_in[11]};
    const float* g [2] = {(const float*)d_in[6],  (const float*)d_in[12]};
    const float* be[2] = {(const float*)d_in[7],  (const float*)d_in[13]};

    const size_t NF = (size_t)GNN_NT * GNN_DOUT;   // 4,096,000 floats
    float* ws   = (float*)d_ws;
    float* X    = ws;             // block0 output / block1 input (and residual)
    float* P    = X + NF;         // xi @ (W1top - W1bot)
    float* Q    = P + NF;         // xj @ W1bot
    float* S    = Q + NF;         // edge-aggregated sums
    float* cnt  = S + NF;         // per-node in-degree (float), contiguous after S
    float* BpA  = cnt + GNN_NT;   // packed (W1top - W1bot)   [<= 128x128]
    float* BpB  = BpA + 128 * GNN_DOUT;   // packed W1bot
    float* Bp2  = BpB + 128 * GNN_DOUT;   // packed W2

    const dim3 egrid(GNN_E * 32 / 256, GNN_BSN);
    const int  zgrid = (int)((NF + GNN_NT + 255) / 256);

    // ---------------- Block 0: din=64 -> dout=128, no residual ----------------
    k_pack_b<true ><<< 64 * GNN_DOUT / 256, 256, 0, stream>>>(W1[0], BpA, 64);
    k_pack_b<false><<< 64 * GNN_DOUT / 256, 256, 0, stream>>>(W1[0] + 64 * GNN_DOUT, BpB, 64);
    k_pack_b<false><<<128 * GNN_DOUT / 256, 256, 0, stream>>>(W2[0], Bp2, 128);
    k_fill_zero<<<zgrid, 256, 0, stream>>>(S, (int)(NF + GNN_NT));   // S and cnt
    k_gemm_pq<64><<<250, 256, 0, stream>>>(x, BpA, BpB, P, Q);
    k_edge<<<egrid, 256, 0, stream>>>(P, Q, b1[0], ei, S, cnt);
    k_gemm_agg_ln<false><<<250, 256, 0, stream>>>(S, Bp2, cnt, b2[0], g[0], be[0], nullptr, X);

    // ---------------- Block 1: din=128 -> dout=128, residual ----------------
    k_pack_b<true ><<<128 * GNN_DOUT / 256, 256, 0, stream>>>(W1[1], BpA, 128);
    k_pack_b<false><<<128 * GNN_DOUT / 256, 256, 0, stream>>>(W1[1] + 128 * GNN_DOUT, BpB, 128);
    k_pack_b<false><<<128 * GNN_DOUT / 256, 256, 0, stream>>>(W2[1], Bp2, 128);
    k_fill_zero<<<zgrid, 256, 0, stream>>>(S, (int)(NF + GNN_NT));
    k_gemm_pq<128><<<250, 256, 0, stream>>>(X, BpA, BpB, P, Q);
    k_edge<<<egrid, 256, 0, stream>>>(P, Q, b1[1], ei, S, cnt);
    k_gemm_agg_ln<true><<<250, 256, 0, stream>>>(S, Bp2, cnt, b2[1], g[1], be[1], X, (float*)d_out);
}